// TopK_34626026340808
// MI455X (gfx1250) — compile-verified
//
#include <hip/hip_runtime.h>
#include <stdint.h>

typedef uint32_t u32;
typedef int   v4i __attribute__((ext_vector_type(4)));
typedef float v4f __attribute__((ext_vector_type(4)));

#define ROWS 8192
#define COLS 24576
#define KSEL 64
#define TPB  1024
#define EPT  (COLS / TPB)        // 24 contiguous elements per thread (for tie ranking)
#define V4PT (COLS / 4 / TPB)    // 6 float4 per thread (interleaved, coalesced)

// LDS layout (dynamic, ~117 KB; 2 blocks / WGP out of 320 KB)
#define OFF_ROW   0
#define OFF_HIST  (OFF_ROW  + COLS * 4)          // 98304
#define OFF_EQ    (OFF_HIST + 4096 * 4)          // +16384
#define OFF_CT    (OFF_EQ   + TPB * 4)           // +4096
#define OFF_BC    (OFF_CT   + 256 * 4)           // +1024
#define LDS_BYTES (OFF_BC   + 64)

typedef __attribute__((address_space(1))) v4i* g4p;   // global 128-bit chunk
typedef __attribute__((address_space(3))) v4i* l4p;   // LDS 128-bit chunk

// monotone key: larger float <-> larger unsigned key
__device__ __forceinline__ u32 sortKey(u32 b) {
  return b ^ ((b & 0x80000000u) ? 0xFFFFFFFFu : 0x80000000u);
}

__device__ __forceinline__ void async_row_load(const float* gp, float* lp) {
#if __has_builtin(__builtin_amdgcn_global_load_async_to_lds_b128)
  __builtin_amdgcn_global_load_async_to_lds_b128(
      (g4p)(v4i*)const_cast<float*>(gp), (l4p)(v4i*)lp, 0, 0);
#else
  u32 ldsoff = (u32)(uintptr_t)(__attribute__((address_space(3))) void*)(void*)lp;
  asm volatile("global_load_async_to_lds_b128 %0, %1, off"
               :: "v"(ldsoff), "v"(gp) : "memory");
#endif
}

__device__ __forceinline__ void wait_async0() {
#if __has_builtin(__builtin_amdgcn_s_wait_asynccnt)
  __builtin_amdgcn_s_wait_asynccnt(0);
#else
  asm volatile("s_wait_asynccnt 0" ::: "memory");
#endif
}

__global__ __launch_bounds__(TPB, 2)
void topk_scatter_kernel(const float* __restrict__ x, float* __restrict__ out) {
  extern __shared__ unsigned char smem[];
  float* s_row  = (float*)(smem + OFF_ROW);   // row staged in LDS
  u32*   s_hist = (u32*)  (smem + OFF_HIST);  // up to 4096 bins
  u32*   s_eq   = (u32*)  (smem + OFF_EQ);    // scan buffer (1024)
  u32*   s_ct   = (u32*)  (smem + OFF_CT);    // 256 chunk totals
  u32*   s_bc   = (u32*)  (smem + OFF_BC);    // {prefix, mask, remaining}

  const int tid = threadIdx.x;
  const int row = blockIdx.x;
  const float* rowp = x + (size_t)row * COLS;

  // ---- 1) async-stage the whole row into LDS (one HBM read per byte) ----
  #pragma unroll
  for (int k = 0; k < V4PT; ++k) {
    int idx4 = tid + k * TPB;                      // coalesced across the wave
    async_row_load(rowp + idx4 * 4, s_row + idx4 * 4);
  }
  if (tid == 0) { s_bc[0] = 0u; s_bc[1] = 0u; s_bc[2] = KSEL; }
  wait_async0();
  __syncthreads();

  // ---- 2) radix-select the K-th largest key: 12 + 12 + 8 bit passes ----
  const int shifts[3] = {20, 8, 0};
  const int nbits [3] = {12, 12, 8};

  for (int p = 0; p < 3; ++p) {
    const int shift = shifts[p];
    const u32 NB    = 1u << nbits[p];
    const u32 bmask = NB - 1u;

    for (u32 i = tid; i < NB; i += TPB) s_hist[i] = 0u;
    __syncthreads();

    const u32 prefix = s_bc[0], msk = s_bc[1], rem = s_bc[2];

    #pragma unroll
    for (int k = 0; k < EPT; ++k) {
      int i = tid + k * TPB;                       // bank-conflict-free LDS reads
      u32 key = sortKey(__float_as_uint(s_row[i]));
      if ((key & msk) == prefix)
        atomicAdd(&s_hist[(key >> shift) & bmask], 1u);
    }
    __syncthreads();

    // chunk totals (256 chunks), then parallel inclusive suffix scan
    const u32 CH = NB >> 8;
    u32 ct = 0;
    if (tid < 256) {
      for (u32 j = 0; j < CH; ++j) ct += s_hist[tid * CH + j];
      s_ct[tid] = ct;
      s_eq[tid] = ct;
    }
    __syncthreads();
    for (u32 off = 1; off < 256; off <<= 1) {
      u32 v = 0;
      if (tid < 256) { v = s_eq[tid]; if (tid + off < 256) v += s_eq[tid + off]; }
      __syncthreads();
      if (tid < 256) s_eq[tid] = v;
      __syncthreads();
    }

    // exactly one chunk satisfies excl < rem <= incl; it refines the bin
    if (tid < 256) {
      u32 incl = s_eq[tid];
      u32 excl = incl - ct;
      if (excl < rem && rem <= incl) {
        u32 above = excl;
        for (int b = (int)(tid * CH + CH) - 1; b >= (int)(tid * CH); --b) {
          u32 c = s_hist[b];
          if (above + c >= rem) {
            s_bc[0] = prefix | ((u32)b << shift);
            s_bc[1] = msk | (bmask << shift);
            s_bc[2] = rem - above;                 // how many == threshold survive
            break;
          }
          above += c;
        }
      }
    }
    __syncthreads();
  }

  const u32 T = s_bc[0];   // exact key of the K-th largest value
  const u32 R = s_bc[2];   // #elements equal to T that are kept (lowest index first)

  // ---- 3) index-ordered ranking of elements equal to T (jax tie-break) ----
  u32 cnt = 0;
  {
    const int base = tid * EPT;
    for (int j = 0; j < EPT; ++j)
      cnt += (sortKey(__float_as_uint(s_row[base + j])) == T) ? 1u : 0u;
    s_eq[tid] = cnt;
    __syncthreads();
    for (u32 off = 1; off < TPB; off <<= 1) {
      u32 v = s_eq[tid];
      if (tid >= off) v += s_eq[tid - off];
      __syncthreads();
      s_eq[tid] = v;
      __syncthreads();
    }
    u32 excl = s_eq[tid] - cnt;
    __syncthreads();
    s_eq[tid] = excl;                              // exclusive index-prefix of equals
    __syncthreads();
  }

  // ---- 4) streaming output: relu(top-k) scattered, zeros elsewhere (NT stores) ----
  float* outp = out + (size_t)row * COLS;
  const v4f* lrow4 = (const v4f*)s_row;
  #pragma unroll
  for (int k = 0; k < V4PT; ++k) {
    int idx4 = tid + k * TPB;
    int e0 = idx4 * 4;
    v4f v = lrow4[idx4];
    float in[4] = {v.x, v.y, v.z, v.w};
    float o[4];
    #pragma unroll
    for (int c = 0; c < 4; ++c) {
      u32 key = sortKey(__float_as_uint(in[c]));
      bool sel = key > T;
      if (!sel && key == T) {                      // rare: rank ties by index
        int e = e0 + c;
        int chunk = e / EPT;
        u32 rank = s_eq[chunk];
        for (int j = chunk * EPT; j < e; ++j)
          rank += (sortKey(__float_as_uint(s_row[j])) == T) ? 1u : 0u;
        sel = rank < R;
      }
      o[c] = sel ? fmaxf(in[c], 0.0f) : 0.0f;
    }
    v4f ov;
    ov.x = o[0]; ov.y = o[1]; ov.z = o[2]; ov.w = o[3];
    __builtin_nontemporal_store(ov, (v4f*)(outp + e0));      // stream past L2
  }
}

extern "C" void kernel_launch(void* const* d_in, const int* in_sizes, int n_in,
                              void* d_out, int out_size, void* d_ws, size_t ws_size,
                              hipStream_t stream) {
  (void)in_sizes; (void)n_in; (void)d_ws; (void)ws_size; (void)out_size;
  const float* x = (const float*)d_in[0];
  float* out = (float*)d_out;

  (void)hipFuncSetAttribute((const void*)topk_scatter_kernel,
                            hipFuncAttributeMaxDynamicSharedMemorySize, (int)LDS_BYTES);

  topk_scatter_kernel<<<dim3(ROWS), dim3(TPB), LDS_BYTES, stream>>>(x, out);
}